// Yolo_Loss_77867757077003
// MI455X (gfx1250) — compile-verified
//
#include <hip/hip_runtime.h>
#include <math.h>

typedef __attribute__((ext_vector_type(2))) float v2f;
typedef __attribute__((ext_vector_type(8))) float v8f;

#define NCLS 20

// ANCHORS / IMG, identical f32 rounding to the reference (np.float32 division)
__constant__ float c_aw[3][3] = {
  {10.f/640.f, 16.f/640.f, 33.f/640.f},
  {30.f/640.f, 62.f/640.f, 59.f/640.f},
  {116.f/640.f,156.f/640.f,373.f/640.f}};
__constant__ float c_ah[3][3] = {
  {13.f/640.f, 30.f/640.f, 23.f/640.f},
  {61.f/640.f, 45.f/640.f,119.f/640.f},
  {90.f/640.f,198.f/640.f,326.f/640.f}};

__device__ __forceinline__ float softplusf(float x) {
  // max(x,0) - x*z + log1p(exp(-|x|)) with z folded out; exact bce identity
  return fmaxf(x, 0.f) + log1pf(expf(-fabsf(x)));
}
__device__ __forceinline__ float sigmoidf(float x) {
  return 1.f / (1.f + expf(-x));
}

__device__ __forceinline__ float iou_cxcywh_dev(float px, float py, float pw, float ph,
                                                float tx, float ty, float tw, float th) {
  float b1x1 = px - pw * 0.5f, b1y1 = py - ph * 0.5f;
  float b1x2 = px + pw * 0.5f, b1y2 = py + ph * 0.5f;
  float b2x1 = tx - tw * 0.5f, b2y1 = ty - th * 0.5f;
  float b2x2 = tx + tw * 0.5f, b2y2 = ty + th * 0.5f;
  float iw = fmaxf(fminf(b1x2, b2x2) - fmaxf(b1x1, b2x1), 0.f);
  float ih = fmaxf(fminf(b1y2, b2y2) - fmaxf(b1y1, b2y1), 0.f);
  float inter = iw * ih;
  float uni = (b1x2 - b1x1) * (b1y2 - b1y1) + (b2x2 - b2x1) * (b2y2 - b2y1) - inter + 1e-7f;
  return inter / uni;
}

// 256 per-thread partials -> single f32 sum using chained V_WMMA_F32_16X16X4_F32
// with B = ones. D row-sums accumulate every A element; total = sum of all 256
// inputs in pure f32 (deterministic fixed order). Valid on lane 0 of wave 0.
__device__ __forceinline__ float block256_sum_wmma(float v, float* sbuf) {
  int t = threadIdx.x;
  sbuf[t] = v;
  __syncthreads();
  float total = 0.f;
  if (t < 32) {               // whole wave 0 -> EXEC all ones for WMMA
    v8f cm = {};
    v2f bm = {1.f, 1.f};
#pragma unroll
    for (int ch = 0; ch < 4; ++ch) {
      v2f am;
      am.x = sbuf[ch * 64 + t];
      am.y = sbuf[ch * 64 + 32 + t];
      cm = __builtin_amdgcn_wmma_f32_16x16x4_f32(false, am, false, bm,
                                                 (short)0, cm, false, false);
    }
    float ls = cm[0] + cm[1] + cm[2] + cm[3] + cm[4] + cm[5] + cm[6] + cm[7];
    // lanes 0-15 hold sum of rows 0..7, lanes 16-31 hold sum of rows 8..15
    total = __shfl(ls, 0, 32) + __shfl(ls, 16, 32);
  }
  return total;
}

// One block per (n, anchor, channel) plane; channels restricted to c in 4..24.
// part[2*plane+0] = obj softplus sum (c==4), part[2*plane+1] = cls softplus sum.
template <int HW>
__global__ void yolo_dense_kernel(const float* __restrict__ p, float* __restrict__ part) {
  constexpr int NV = HW / 4;              // float4 elements per plane (HW % 4 == 0)
  constexpr int K  = (NV + 255) / 256;
  __shared__ float sbuf[256];

  int plane = blockIdx.x;                 // [0, N*63)
  int n   = plane / 63;
  int rem = plane - n * 63;
  int a   = rem / 21;
  int c   = rem - a * 21;                 // 0 -> obj (ch 4), 1..20 -> cls (ch 5..24)
  const float4* base4 =
      (const float4*)(p + (size_t)(n * 75 + a * 25 + 4 + c) * HW);

  int t = threadIdx.x;
  float acc = 0.f;
#pragma unroll
  for (int k = 0; k < K; ++k) {
    int i4 = t + k * 256;
    if ((NV % 256 == 0) || (i4 < NV)) {
      float4 x = base4[i4];               // global_load_b128, fully coalesced
      acc += softplusf(x.x) + softplusf(x.y) + softplusf(x.z) + softplusf(x.w);
    }
  }

  float total = block256_sum_wmma(acc, sbuf);
  if (t == 0) {
    part[2 * plane + 0] = (c == 0) ? total : 0.f;
    part[2 * plane + 1] = (c == 0) ? 0.f : total;
  }
}

// One block per scale, one thread per image n. Replicates build_target's
// sequential scan semantics: a cell's value comes from the LAST valid target
// mapping to it; npos counts distinct written cells.
__global__ void yolo_sparse_kernel(const float* __restrict__ p0, const float* __restrict__ p1,
                                   const float* __restrict__ p2, const float* __restrict__ tgt,
                                   float* __restrict__ sp, int N, int T) {
  int s = blockIdx.x;
  const float* p = (s == 0) ? p0 : (s == 1 ? p1 : p2);
  float stride = (float)(8 << s);
  int W = 640 >> (3 + s);                 // 80, 40, 20
  int H = W, HW = H * W;
  int n = threadIdx.x;                    // blockDim.x == N

  float box_sum = 0.f, nposf = 0.f, obj_corr = 0.f, cls_corr = 0.f;
  int key[64];
  int Tc = (T > 64) ? 64 : T;

  // pass 1: compute cell keys (anchor,gy,gx) for valid targets
  for (int t = 0; t < Tc; ++t) {
    const float* tr = tgt + ((size_t)n * T + t) * 6;
    float x1 = tr[0], y1 = tr[1], x2 = tr[2], y2 = tr[3];
    int k = -1;
    if ((x2 > x1) && (y2 > y1)) {
      float cx = (x1 + x2) * 0.5f, cy = (y1 + y2) * 0.5f;
      float bw = x2 - x1, bh = y2 - y1;
      float fx = fminf(fmaxf(cx / stride, 0.f), (float)(W - 1));
      float fy = fminf(fmaxf(cy / stride, 0.f), (float)(H - 1));
      int gx = (int)fx, gy = (int)fy;
      float bestv = -3.0e38f;
      int best = 0;
#pragma unroll
      for (int a = 0; a < 3; ++a) {       // first-max argmax like jnp.argmax
        float r0 = bw / c_aw[s][a];
        float r1 = bh / c_ah[s][a];
        float v = fminf(r0, 1.f / (r0 + 1e-6f)) * fminf(r1, 1.f / (r1 + 1e-6f));
        if (v > bestv) { bestv = v; best = a; }
      }
      k = (best * H + gy) * W + gx;
    }
    key[t] = k;
  }

  // pass 2: winners = valid targets with no later valid target on same cell
  for (int t = 0; t < Tc; ++t) {
    int k = key[t];
    if (k < 0) continue;
    bool winner = true;
    for (int t2 = t + 1; t2 < Tc; ++t2)
      if (key[t2] == k) { winner = false; break; }
    if (!winner) continue;

    const float* tr = tgt + ((size_t)n * T + t) * 6;
    float x1 = tr[0], y1 = tr[1], x2 = tr[2], y2 = tr[3], cls = tr[5];
    float cx = (x1 + x2) * 0.5f, cy = (y1 + y2) * 0.5f;
    float bw = x2 - x1, bh = y2 - y1;

    int best = k / HW;
    int rem = k - best * HW;
    int gy = rem / W;
    int gx = rem - gy * W;
    float dx = cx / stride - (float)gx;
    float dy = cy / stride - (float)gy;
    float aw = c_aw[s][best], ah = c_ah[s][best];

    size_t base = (size_t)(n * 75 + best * 25) * HW + (size_t)gy * W + gx;
    float ptx  = p[base];
    float pty  = p[base + (size_t)HW];
    float ptw  = p[base + (size_t)2 * HW];
    float pth  = p[base + (size_t)3 * HW];
    float pobj = p[base + (size_t)4 * HW];

    int cls_i = (int)cls;                                 // trunc, like astype(int32)
    bool cls_ok = (cls_i >= 0) && (cls_i < NCLS);
    int cc = cls_i < 0 ? 0 : (cls_i > NCLS - 1 ? NCLS - 1 : cls_i);
    float pcls = cls_ok ? p[base + (size_t)(5 + cc) * HW] : 0.f;

    float px = (2.f * sigmoidf(ptx) - 0.5f + (float)gx) * stride;
    float py = (2.f * sigmoidf(pty) - 0.5f + (float)gy) * stride;
    float swv = 2.f * sigmoidf(ptw);
    float shv = 2.f * sigmoidf(pth);
    float pw = swv * swv * aw * stride;
    float ph = shv * shv * ah * stride;

    float txp = (dx + 0.5f + (float)gx) * stride;
    float typ = (dy + 0.5f + (float)gy) * stride;
    float ttw = bw / aw, tth = bh / ah;
    float tw = ttw * ttw * aw;                            // reference has no stride here
    float th = tth * tth * ah;

    float iou = iou_cxcywh_dev(px, py, pw, ph, txp, typ, tw, th);
    box_sum  += 1.f - iou;
    nposf    += 1.f;
    obj_corr += pobj;                                     // z_obj = 1
    cls_corr += pcls;                                     // sum_c onehot_c * x_c
  }

  __shared__ float red[4][256];
  red[0][n] = box_sum; red[1][n] = nposf; red[2][n] = obj_corr; red[3][n] = cls_corr;
  __syncthreads();
  if (n == 0) {
    float a0 = 0.f, a1 = 0.f, a2 = 0.f, a3 = 0.f;
    for (int i = 0; i < N; ++i) {
      a0 += red[0][i]; a1 += red[1][i]; a2 += red[2][i]; a3 += red[3][i];
    }
    sp[s * 4 + 0] = a0; sp[s * 4 + 1] = a1; sp[s * 4 + 2] = a2; sp[s * 4 + 3] = a3;
  }
}

__global__ void yolo_compose_kernel(const float* __restrict__ part, const float* __restrict__ sp,
                                    float* __restrict__ out, int planes, int N) {
  __shared__ float red[256];
  __shared__ float sums[6];
  int t = threadIdx.x;
  for (int kk = 0; kk < 6; ++kk) {        // {scale0..2} x {obj,cls}
    int s = kk >> 1, j = kk & 1;
    const float* arr = part + (size_t)s * planes * 2;
    float acc = 0.f;
    for (int i = t; i < planes; i += 256) acc += arr[2 * i + j];
    red[t] = acc;
    __syncthreads();
    for (int off = 128; off > 0; off >>= 1) {
      if (t < off) red[t] += red[t + off];
      __syncthreads();
    }
    if (t == 0) sums[kk] = red[0];
    __syncthreads();
  }
  if (t == 0) {
    const int HWs[3] = {6400, 1600, 400};
    float total = 0.f, lb = 0.f, lo = 0.f, lc = 0.f;
    for (int s = 0; s < 3; ++s) {
      float M = (float)((long long)N * 3 * HWs[s]);
      float box_sum  = sp[s * 4 + 0];
      float npos     = sp[s * 4 + 1];
      float obj_corr = sp[s * 4 + 2];
      float cls_corr = sp[s * 4 + 3];
      float lobj = (sums[s * 2 + 0] - obj_corr) / M;
      float lcls = (npos > 0.f) ? (sums[s * 2 + 1] - cls_corr) / (M * (float)NCLS) : 0.f;
      float lbox = box_sum / fmaxf(npos, 1.f);
      total += 0.05f * lbox + 1.0f * lobj + 0.5f * lcls;
      lb += lbox; lo += lobj; lc += lcls;
    }
    out[0] = total; out[1] = lb; out[2] = lo; out[3] = lc;
  }
}

extern "C" void kernel_launch(void* const* d_in, const int* in_sizes, int n_in,
                              void* d_out, int out_size, void* d_ws, size_t ws_size,
                              hipStream_t stream) {
  const float* p0 = (const float*)d_in[0];
  const float* p1 = (const float*)d_in[1];
  const float* p2 = (const float*)d_in[2];
  const float* tg = (const float*)d_in[3];

  int N = in_sizes[0] / (75 * 80 * 80);   // 32
  int T = in_sizes[3] / (N * 6);          // 60
  int planes = N * 63;                    // N * A * 21 channels

  float* ws = (float*)d_ws;
  float* part = ws;                       // [3][planes][2] floats
  float* sp   = ws + (size_t)3 * planes * 2;  // [3][4] floats

  yolo_dense_kernel<6400><<<planes, 256, 0, stream>>>(p0, part);
  yolo_dense_kernel<1600><<<planes, 256, 0, stream>>>(p1, part + (size_t)planes * 2);
  yolo_dense_kernel<400><<<planes, 256, 0, stream>>>(p2, part + (size_t)planes * 4);
  yolo_sparse_kernel<<<3, N, 0, stream>>>(p0, p1, p2, tg, sp, N, T);
  yolo_compose_kernel<<<1, 256, 0, stream>>>(part, sp, (float*)d_out, planes, N);
}